// LoopDecoder_12120397710137
// MI455X (gfx1250) — compile-verified
//
#include <hip/hip_runtime.h>

#define BB 8
#define NN 1000
#define GG 1000
#define EE 128
#define HH 8
#define DHD 16
#define KNB 16
#define NT 63   /* ceil(1000/16) */

typedef __attribute__((ext_vector_type(2))) float v2f;
typedef __attribute__((ext_vector_type(8))) float v8f;

__device__ __forceinline__ v8f wmma4(v2f a, v2f b, v8f c) {
  // V_WMMA_F32_16X16X4_F32 : D = A(16x4) * B(4x16) + C(16x16), all f32
  return __builtin_amdgcn_wmma_f32_16x16x4_f32(
      /*neg_a=*/false, a, /*neg_b=*/false, b,
      /*c_mod=*/(short)0, c, /*reuse_a=*/false, /*reuse_b=*/false);
}

// ---------------- graph mean + q_graph (tiny) ----------------
__global__ void mean_kernel(const float* __restrict__ emb, float* __restrict__ meanv) {
  int b = blockIdx.x, e = threadIdx.x;
  const float* p = emb + (size_t)b * NN * EE + e;
  float s = 0.f;
  for (int n = 0; n < NN; ++n) s += p[(size_t)n * EE];
  meanv[b * EE + e] = s * (1.0f / NN);
}

__global__ void qgraph_kernel(const float* __restrict__ meanv,
                              const float* __restrict__ Wq,
                              float* __restrict__ qg) {
  int b = blockIdx.x, e = threadIdx.x;
  const float* m = meanv + b * EE;
  const float* w = Wq + (size_t)e * EE;
  float s = 0.f;
  for (int k = 0; k < EE; ++k) s += m[k] * w[k];
  qg[b * EE + e] = s;
}

// ---------------- generic projection: Y = X @ W^T  (M = gridDim.x*16 rows) ----------------
__global__ void proj_kernel(const float* __restrict__ X, const float* __restrict__ W,
                            float* __restrict__ Y) {
  const int lane = threadIdx.x & 31;
  const int wave = threadIdx.x >> 5;      // column tile (8 tiles cover EE=128)
  const int mt   = blockIdx.x;
  const int l16  = lane & 15;
  const int half = lane >> 4;
  const int arow = mt * 16 + l16;
  const int bcol = wave * 16 + l16;
  const float* ap = X + (size_t)arow * EE;
  const float* bp = W + (size_t)bcol * EE;   // B[k][n] = W[n][k]
  v8f c = {};
#pragma unroll
  for (int kk = 0; kk < EE / 4; ++kk) {
    int k0 = kk * 4 + half * 2;
    v2f a = *(const v2f*)(ap + k0);
    v2f b = *(const v2f*)(bp + k0);
    c = wmma4(a, b, c);
  }
#pragma unroll
  for (int r = 0; r < 8; ++r) {
    int row = mt * 16 + r + half * 8;
    Y[(size_t)row * EE + bcol] = c[r];
  }
}

// ---------------- glimpse_q = gather(emb,last) @ (Wqf+Wql)^T + q_graph[b] ----------------
__global__ void glimpseq_kernel(const float* __restrict__ emb,
                                const int* __restrict__ last_node,
                                const float* __restrict__ Wqf,
                                const float* __restrict__ Wql,
                                const float* __restrict__ qg,
                                float* __restrict__ Y) {
  const int lane = threadIdx.x & 31;
  const int wave = threadIdx.x >> 5;
  const int mt   = blockIdx.x;
  const int l16  = lane & 15;
  const int half = lane >> 4;
  const int m    = mt * 16 + l16;            // flat (b,g)
  const int b    = m / GG;
  const int node = last_node[m];
  const int bcol = wave * 16 + l16;
  const float* ap = emb + ((size_t)b * NN + node) * EE;
  const float* bf = Wqf + (size_t)bcol * EE;
  const float* bl = Wql + (size_t)bcol * EE;
  v8f c = {};
#pragma unroll
  for (int kk = 0; kk < EE / 4; ++kk) {
    int k0 = kk * 4 + half * 2;
    v2f a  = *(const v2f*)(ap + k0);
    v2f w0 = *(const v2f*)(bf + k0);
    v2f w1 = *(const v2f*)(bl + k0);
    v2f w; w.x = w0.x + w1.x; w.y = w0.y + w1.y;
    c = wmma4(a, w, c);
  }
#pragma unroll
  for (int r = 0; r < 8; ++r) {
    int row = mt * 16 + r + half * 8;
    int rb  = row / GG;
    Y[(size_t)row * EE + bcol] = c[r] + qg[rb * EE + bcol];
  }
}

// ---------------- final_q = out @ Wcomb^T + b_comb ----------------
__global__ void comb_kernel(const float* __restrict__ X, const float* __restrict__ W,
                            const float* __restrict__ bias, float* __restrict__ Y) {
  const int lane = threadIdx.x & 31;
  const int wave = threadIdx.x >> 5;
  const int mt   = blockIdx.x;
  const int l16  = lane & 15;
  const int half = lane >> 4;
  const int arow = mt * 16 + l16;
  const int bcol = wave * 16 + l16;
  const float* ap = X + (size_t)arow * EE;
  const float* bp = W + (size_t)bcol * EE;
  v8f c = {};
#pragma unroll
  for (int kk = 0; kk < EE / 4; ++kk) {
    int k0 = kk * 4 + half * 2;
    v2f a = *(const v2f*)(ap + k0);
    v2f b = *(const v2f*)(bp + k0);
    c = wmma4(a, b, c);
  }
  const float bv = bias[bcol];
#pragma unroll
  for (int r = 0; r < 8; ++r) {
    int row = mt * 16 + r + half * 8;
    Y[(size_t)row * EE + bcol] = c[r] + bv;
  }
}

// ---------------- KNN top-16 per (b,g) ----------------
__global__ void knn_kernel(const int* __restrict__ last_node,
                           const float* __restrict__ coords,
                           const float* __restrict__ mask,
                           int* __restrict__ knn_out) {
  const int tid = blockIdx.x * blockDim.x + threadIdx.x;
  if (tid >= BB * GG) return;
  const int b = tid / GG;
  const int node = last_node[tid];
  const float cx = coords[((size_t)b * NN + node) * 2 + 0];
  const float cy = coords[((size_t)b * NN + node) * 2 + 1];
  const float INF = __builtin_inff();
  float bd[KNB]; int bi[KNB];
  for (int j = 0; j < KNB; ++j) { bd[j] = INF; bi[j] = 0; }
  const float* cp = coords + (size_t)b * NN * 2;
  const float* mp = mask + (size_t)tid * NN;
  for (int n = 0; n < NN; ++n) {
    float dx = cp[n * 2] - cx, dy = cp[n * 2 + 1] - cy;
    float d = sqrtf(fmaxf(dx * dx + dy * dy, 0.f));
    if (mp[n] == -INF) d = INF;
    if (d < bd[KNB - 1]) {                 // stable insertion (ties keep lower index)
      int j = KNB - 1;
      while (j > 0 && bd[j - 1] > d) { bd[j] = bd[j - 1]; bi[j] = bi[j - 1]; --j; }
      bd[j] = d; bi[j] = n;
    }
  }
  for (int j = 0; j < KNB; ++j) knn_out[tid * KNB + j] = bi[j];
}

// ---------------- 16-neighbor glimpse attention, one thread per (b,g,h) ----------------
__global__ void attn_kernel(const float* __restrict__ gq,
                            const float* __restrict__ Kb,
                            const float* __restrict__ Vb,
                            const int* __restrict__ knn,
                            float* __restrict__ outp) {
  const int tid = blockIdx.x * blockDim.x + threadIdx.x;
  if (tid >= BB * GG * HH) return;
  const int h = tid % HH;
  const int bg = tid / HH;
  const int b = bg / GG;
  const float* q = gq + (size_t)bg * EE + h * DHD;
  float qv[DHD];
#pragma unroll
  for (int d = 0; d < DHD; ++d) qv[d] = q[d];
  int ni[KNB]; float sc[KNB];
  float mx = -__builtin_inff();
#pragma unroll
  for (int j = 0; j < KNB; ++j) {
    int n = knn[bg * KNB + j];
    ni[j] = n;
    const float* k = Kb + ((size_t)b * NN + n) * EE + h * DHD;
    float s = 0.f;
#pragma unroll
    for (int d = 0; d < DHD; ++d) s += qv[d] * k[d];
    s *= 0.25f;                      // 1/sqrt(DH)
    sc[j] = s;
    mx = fmaxf(mx, s);
  }
  float se = 0.f;
#pragma unroll
  for (int j = 0; j < KNB; ++j) { sc[j] = __expf(sc[j] - mx); se += sc[j]; }
  const float inv = 1.0f / se;
  float ov[DHD];
#pragma unroll
  for (int d = 0; d < DHD; ++d) ov[d] = 0.f;
#pragma unroll
  for (int j = 0; j < KNB; ++j) {
    const float* v = Vb + ((size_t)b * NN + ni[j]) * EE + h * DHD;
    float p = sc[j] * inv;
#pragma unroll
    for (int d = 0; d < DHD; ++d) ov[d] += p * v[d];
  }
  float* o = outp + (size_t)bg * EE + h * DHD;
#pragma unroll
  for (int d = 0; d < DHD; ++d) o[d] = ov[d];
}

// ---------------- fused score GEMM + tanh-clip + row softmax ----------------
__global__ void score_softmax_kernel(const float* __restrict__ fq,
                                     const float* __restrict__ emb,
                                     const float* __restrict__ mask,
                                     float* __restrict__ out) {
  const int lane = threadIdx.x & 31;
  const int wave = threadIdx.x >> 5;
  const int gt = blockIdx.x;         // 16-row tile of G
  const int b  = blockIdx.y;
  const int l16  = lane & 15;
  const int half = lane >> 4;
  const float INF = __builtin_inff();
  const float scale = 0.08838834764831845f;  // 1/sqrt(128)

  __shared__ float sm_m[8][16];
  __shared__ float sm_l[8][16];
  __shared__ float fin_m[16];
  __shared__ float fin_il[16];

  const int arow = gt * 16 + l16;
  const int arc = (arow < GG) ? arow : (GG - 1);
  const float* ap = fq + ((size_t)b * GG + arc) * EE;

  // A fragment (16 rows of final_q, full K=128) is invariant across the
  // column sweep: keep it in registers (64 floats/lane).
  v2f afrag[EE / 4];
#pragma unroll
  for (int kk = 0; kk < EE / 4; ++kk)
    afrag[kk] = *(const v2f*)(ap + kk * 4 + half * 2);

  float m_run[8], l_run[8];
#pragma unroll
  for (int r = 0; r < 8; ++r) { m_run[r] = -INF; l_run[r] = 0.f; }

  for (int nt = wave; nt < NT; nt += 8) {           // uniform per wave
    const int ncol = nt * 16 + l16;
    const int ncc = (ncol < NN) ? ncol : (NN - 1);
    const float* bp = emb + ((size_t)b * NN + ncc) * EE;

    // prefetch next column tile's emb row while this tile computes
    {
      const int pcol = ncol + 8 * 16;
      const int pcc = (pcol < NN) ? pcol : (NN - 1);
      __builtin_prefetch(emb + ((size_t)b * NN + pcc) * EE, 0, 3);
    }

    v8f c = {};
#pragma unroll
    for (int kk = 0; kk < EE / 4; ++kk) {
      v2f bb = *(const v2f*)(bp + kk * 4 + half * 2);
      c = wmma4(afrag[kk], bb, c);
    }
#pragma unroll
    for (int r = 0; r < 8; ++r) {
      const int g = gt * 16 + r + half * 8;
      float s;
      if (g < GG && ncol < NN) {
        size_t idx = ((size_t)(b * GG + g)) * NN + ncol;
        s = 10.f * tanhf(c[r] * scale) + mask[idx];
        out[idx] = s;                               // raw masked score (pass 1)
      } else {
        s = -INF;
      }
      float mo = m_run[r];
      float mn = fmaxf(mo, s);
      if (mn > -INF) {
        l_run[r] = l_run[r] * __expf(mo - mn) + __expf(s - mn);
        m_run[r] = mn;
      }
    }
  }

  // reduce (m,l) across the 16 lanes that share each row
#pragma unroll
  for (int r = 0; r < 8; ++r) {
#pragma unroll
    for (int off = 8; off >= 1; off >>= 1) {
      float mo = __shfl_xor(m_run[r], off, 16);
      float lo = __shfl_xor(l_run[r], off, 16);
      float mn = fmaxf(m_run[r], mo);
      if (mn > -INF) {
        l_run[r] = l_run[r] * __expf(m_run[r] - mn) + lo * __expf(mo - mn);
        m_run[r] = mn;
      }
    }
  }
  if (l16 == 0) {
#pragma unroll
    for (int r = 0; r < 8; ++r) {
      sm_m[wave][r + half * 8] = m_run[r];
      sm_l[wave][r + half * 8] = l_run[r];
    }
  }
  __syncthreads();
  if (threadIdx.x < 16) {
    float M = -INF, L = 0.f;
    for (int w = 0; w < 8; ++w) {
      float mo = sm_m[w][threadIdx.x], lo = sm_l[w][threadIdx.x];
      float mn = fmaxf(M, mo);
      if (mn > -INF) {
        L = L * __expf(M - mn) + lo * __expf(mo - mn);
        M = mn;
      }
    }
    fin_m[threadIdx.x]  = M;
    fin_il[threadIdx.x] = (L > 0.f) ? (1.0f / L) : 0.f;
  }
  __syncthreads();

  // pass 2: each lane re-reads exactly the addresses it wrote (same-wave ordered)
  for (int nt = wave; nt < NT; nt += 8) {
#pragma unroll
    for (int r = 0; r < 8; ++r) {
      const int g = gt * 16 + r + half * 8;
      const int ncol = nt * 16 + l16;
      if (g < GG && ncol < NN) {
        size_t idx = ((size_t)(b * GG + g)) * NN + ncol;
        float s = out[idx];
        int rl = r + half * 8;
        out[idx] = __expf(s - fin_m[rl]) * fin_il[rl];
      }
    }
  }
}

extern "C" void kernel_launch(void* const* d_in, const int* in_sizes, int n_in,
                              void* d_out, int out_size, void* d_ws, size_t ws_size,
                              hipStream_t stream) {
  (void)in_sizes; (void)n_in; (void)out_size; (void)ws_size;
  const int*   last_node = (const int*)d_in[0];
  const float* coords    = (const float*)d_in[1];
  const float* emb       = (const float*)d_in[2];
  const float* gmask     = (const float*)d_in[3];
  const float* Wq_graph  = (const float*)d_in[4];
  const float* Wq_first  = (const float*)d_in[5];
  const float* Wq_last   = (const float*)d_in[6];
  const float* Wk        = (const float*)d_in[7];
  const float* Wv        = (const float*)d_in[8];
  const float* W_comb    = (const float*)d_in[9];
  const float* b_comb    = (const float*)d_in[10];
  float* out = (float*)d_out;

  float* ws    = (float*)d_ws;
  float* meanv = ws;                       // B*E
  float* qg    = meanv + BB * EE;          // B*E
  float* Kbuf  = qg + BB * EE;             // B*N*E
  float* Vbuf  = Kbuf + (size_t)BB * NN * EE;
  float* gqb   = Vbuf + (size_t)BB * NN * EE;   // B*G*E
  float* aout  = gqb + (size_t)BB * GG * EE;    // B*G*E
  float* fqb   = aout + (size_t)BB * GG * EE;   // B*G*E
  int*   knn   = (int*)(fqb + (size_t)BB * GG * EE);  // B*G*16

  mean_kernel<<<BB, EE, 0, stream>>>(emb, meanv);
  qgraph_kernel<<<BB, EE, 0, stream>>>(meanv, Wq_graph, qg);
  proj_kernel<<<(BB * NN) / 16, 256, 0, stream>>>(emb, Wk, Kbuf);
  proj_kernel<<<(BB * NN) / 16, 256, 0, stream>>>(emb, Wv, Vbuf);
  glimpseq_kernel<<<(BB * GG) / 16, 256, 0, stream>>>(emb, last_node, Wq_first, Wq_last, qg, gqb);
  knn_kernel<<<(BB * GG + 255) / 256, 256, 0, stream>>>(last_node, coords, gmask, knn);
  attn_kernel<<<(BB * GG * HH + 255) / 256, 256, 0, stream>>>(gqb, Kbuf, Vbuf, knn, aout);
  comb_kernel<<<(BB * GG) / 16, 256, 0, stream>>>(aout, W_comb, b_comb, fqb);
  score_softmax_kernel<<<dim3(NT, BB), 256, 0, stream>>>(fqb, emb, gmask, out);
}